// YinYangAlphaGridMask_73349451481882
// MI455X (gfx1250) — compile-verified
//
#include <hip/hip_runtime.h>
#include <cstdint>

#define GRID_R 128
#define BLK 256
#define FLOATS_PER_SAMPLE 7
#define BLOCK_FLOATS (BLK * FLOATS_PER_SAMPLE)   // 1792 floats
#define BLOCK_BYTES  (BLOCK_FLOATS * 4)          // 7168 bytes
#define XFERS_B128   (BLOCK_BYTES / 16)          // 448 b128 lane-transfers

// ---------------- CDNA5 async global->LDS staging helpers ----------------
typedef __attribute__((__vector_size__(16))) int v4i;

__device__ __forceinline__ void async_load_b128(const void* gptr, void* lptr) {
#if defined(__gfx1250__) && __has_builtin(__builtin_amdgcn_global_load_async_to_lds_b128)
  typedef __attribute__((address_space(1))) v4i gv4i;   // global v4i
  typedef __attribute__((address_space(3))) v4i lv4i;   // LDS v4i
  __builtin_amdgcn_global_load_async_to_lds_b128(
      (gv4i*)(uintptr_t)gptr,
      (lv4i*)(uint32_t)(uintptr_t)lptr,
      0 /*imm offset*/, 0 /*imm cpol*/);
#else
  // ISA 15.18.3 opcode 98: GLOBAL_LOAD_ASYNC_TO_LDS_B128, GV mode.
  // vdst = LDS byte address VGPR, vaddr = 64-bit global address VGPR pair.
  uint32_t l = (uint32_t)(uintptr_t)lptr;
  uint64_t g = (uint64_t)(uintptr_t)gptr;
  asm volatile("global_load_async_to_lds_b128 %0, %1, off"
               :: "v"(l), "v"(g) : "memory");
#endif
}

__device__ __forceinline__ void wait_async0() {
#if defined(__gfx1250__) && __has_builtin(__builtin_amdgcn_s_wait_asynccnt)
  __builtin_amdgcn_s_wait_asynccnt(0);
#else
  asm volatile("s_wait_asynccnt 0" ::: "memory");
#endif
}

// -------------------------------------------------------------------------
__global__ void __launch_bounds__(BLK)
yinyang_trilinear_kernel(const float* __restrict__ samples,  // [n,7]
                         const float* __restrict__ vol_yin,  // [128,128,128]
                         const float* __restrict__ vol_yang, // [128,128,128]
                         float* __restrict__ out,            // [n]
                         int n) {
  __shared__ float smem[BLOCK_FLOATS];

  const int t    = threadIdx.x;
  const int base = blockIdx.x * BLK;
  float s[FLOATS_PER_SAMPLE];

  if (base + BLK <= n) {
    // Full block: async-stage 256 samples (7168 B) into LDS via b128 lane DMAs.
    const char* gbase = (const char*)(samples) + (size_t)base * (FLOATS_PER_SAMPLE * 4);
    char*       lbase = (char*)smem;
    async_load_b128(gbase + (size_t)t * 16, lbase + (size_t)t * 16);
    if (t < XFERS_B128 - BLK) {  // t < 192: second transfer covers bytes [4096,7168)
      async_load_b128(gbase + (size_t)(BLK + t) * 16, lbase + (size_t)(BLK + t) * 16);
    }
    wait_async0();       // drain this wave's ASYNCcnt
    __syncthreads();     // make every wave's staged data visible block-wide
#pragma unroll
    for (int k = 0; k < FLOATS_PER_SAMPLE; ++k)
      s[k] = smem[t * FLOATS_PER_SAMPLE + k];   // stride-7 dwords: conflict-free
  } else {
    const int i = base + t;
    if (i >= n) return;
#pragma unroll
    for (int k = 0; k < FLOATS_PER_SAMPLE; ++k)
      s[k] = samples[(size_t)i * FLOATS_PER_SAMPLE + k];
  }

  const int i = base + t;
  if (i >= n) return;

  // Route to yin (flag==0) or yang grid; do ONE trilinear sample (8 gathers).
  const bool yin = (s[6] == 0.0f);
  const float x = yin ? s[0] : s[3];
  const float y = yin ? s[1] : s[4];
  const float z = yin ? s[2] : s[5];
  const float* __restrict__ vol = yin ? vol_yin : vol_yang;

  // align_corners=True unnormalize: [-1,1] -> [0, 127]
  const float scale = 0.5f * (float)(GRID_R - 1);
  const float u = (x + 1.0f) * scale;
  const float v = (y + 1.0f) * scale;
  const float w = (z + 1.0f) * scale;

  int x0 = min(max((int)floorf(u), 0), GRID_R - 1);
  int y0 = min(max((int)floorf(v), 0), GRID_R - 1);
  int z0 = min(max((int)floorf(w), 0), GRID_R - 1);
  const int x1 = min(x0 + 1, GRID_R - 1);
  const int y1 = min(y0 + 1, GRID_R - 1);
  const int z1 = min(z0 + 1, GRID_R - 1);

  const float wx = u - (float)x0;
  const float wy = v - (float)y0;
  const float wz = w - (float)z0;

  const int zy00 = (z0 << 14) + (y0 << 7);
  const int zy01 = (z0 << 14) + (y1 << 7);
  const int zy10 = (z1 << 14) + (y0 << 7);
  const int zy11 = (z1 << 14) + (y1 << 7);

  const float c000 = vol[zy00 + x0];
  const float c001 = vol[zy00 + x1];
  const float c010 = vol[zy01 + x0];
  const float c011 = vol[zy01 + x1];
  const float c100 = vol[zy10 + x0];
  const float c101 = vol[zy10 + x1];
  const float c110 = vol[zy11 + x0];
  const float c111 = vol[zy11 + x1];

  const float omx = 1.0f - wx;
  const float c00 = c000 * omx + c001 * wx;
  const float c01 = c010 * omx + c011 * wx;
  const float c10 = c100 * omx + c101 * wx;
  const float c11 = c110 * omx + c111 * wx;
  const float omy = 1.0f - wy;
  const float c0  = c00 * omy + c01 * wy;
  const float c1  = c10 * omy + c11 * wy;
  out[i] = c0 * (1.0f - wz) + c1 * wz;
}

extern "C" void kernel_launch(void* const* d_in, const int* in_sizes, int n_in,
                              void* d_out, int out_size, void* d_ws, size_t ws_size,
                              hipStream_t stream) {
  const float* samples  = (const float*)d_in[0];  // [N,7] float32
  const float* vol_yin  = (const float*)d_in[1];  // [128^3] float32
  const float* vol_yang = (const float*)d_in[2];  // [128^3] float32
  float*       out      = (float*)d_out;          // [N] float32

  const int n = in_sizes[0] / FLOATS_PER_SAMPLE;  // 4194304
  const int blocks = (n + BLK - 1) / BLK;         // 16384
  hipLaunchKernelGGL(yinyang_trilinear_kernel, dim3(blocks), dim3(BLK), 0, stream,
                     samples, vol_yin, vol_yang, out, n);
}